// _CompiledStepBody_67233418052193
// MI455X (gfx1250) — compile-verified
//
#include <hip/hip_runtime.h>

// ---------------------------------------------------------------------------
// econ_step collapsed to the output-relevant dataflow (gdp, aec only).
// Streaming, bandwidth-bound: ~171 MB read / 1 MB write per call.
// VALU-trimmed: fast rcp instead of IEEE division chains; register-latched
// outputs (2 coalesced stores per wave instead of 2 per row).
// ---------------------------------------------------------------------------

#define DT_F      0.1f
#define EPS_F     1e-6f
#define JJ        64
#define MM        32
#define TILE_R    128                 // rows per block
#define NTHREADS  256                 // 8 wave32s
#define ROWS_PER_WAVE (TILE_R / 8)    // 16
#define LDS_STRIDE (TILE_R + 1)       // 129 -> conflict-free column reads

typedef __attribute__((ext_vector_type(2))) float v2f;
typedef __attribute__((ext_vector_type(4))) float v4f;
typedef __attribute__((ext_vector_type(8))) float v8f;

// v_rcp_f32 (1 ulp) + one Newton-Raphson step -> ~0.5 ulp reciprocal,
// 3-4 VALU ops instead of the ~11-op IEEE division chain.
__device__ __forceinline__ float fast_rcp(float x) {
#if __has_builtin(__builtin_amdgcn_rcpf)
  float r = __builtin_amdgcn_rcpf(x);
  r = __builtin_fmaf(__builtin_fmaf(-x, r, 1.0f), r, r);
  return r;
#else
  return 1.0f / x;
#endif
}

// Exact fp32 32-lane sum via V_WMMA_F32_16X16X4_F32 with A = ones(16x4):
//   D[m][n] = sum_k B[k][n]; column n of B occupies lanes {n, n+16} of B's
//   two VGPRs under any documented striping. With B0=x, B1=shfl_xor(x,8):
//   D0[lane] = x[n]+x[n^8]+x[n^16]+x[n^24] (n=lane&15); 3 xor-steps finish.
// Result is broadcast to all 32 lanes.
__device__ __forceinline__ float wave_sum32(float x) {
#if __has_builtin(__builtin_amdgcn_wmma_f32_16x16x4_f32)
  v2f a; a.x = 1.0f; a.y = 1.0f;                 // A = ones (layout-agnostic)
  v2f b; b.x = x;    b.y = __shfl_xor(x, 8, 32); // B VGPR0, VGPR1
  v8f c = {};
  c = __builtin_amdgcn_wmma_f32_16x16x4_f32(false, a, false, b,
                                            (short)0, c, false, false);
  float y = c[0];                                // 4-lane coset sum, all lanes
  y += __shfl_xor(y, 4, 32);
  y += __shfl_xor(y, 2, 32);
  y += __shfl_xor(y, 1, 32);
  return y;
#else
  x += __shfl_xor(x, 16, 32);
  x += __shfl_xor(x, 8, 32);
  x += __shfl_xor(x, 4, 32);
  x += __shfl_xor(x, 2, 32);
  x += __shfl_xor(x, 1, 32);
  return x;
#endif
}

__device__ __forceinline__ float clampf(float v, float lo, float hi) {
  return fminf(fmaxf(v, lo), hi);
}

__global__ __launch_bounds__(NTHREADS) void econ_gdp_aec_kernel(
    const float* __restrict__ need_prev,
    const float* __restrict__ inventory,
    const float* __restrict__ endowment,
    const float* __restrict__ price,        // J x R (transposed access)
    const float* __restrict__ pool_atp,
    const float* __restrict__ pool_adp,
    const float* __restrict__ pool_amp,
    const float* __restrict__ greed,
    const float* __restrict__ innovation,
    const float* __restrict__ extraction_eff,
    const float* __restrict__ deposits,
    const float* __restrict__ exergy_price,
    const float* __restrict__ bases,
    const float* __restrict__ scales,
    float* __restrict__ out_gdp,
    float* __restrict__ out_aec,
    int R)
{
  __shared__ float lds_price[JJ * LDS_STRIDE];   // 33 KB

  const int t  = threadIdx.x;
  const int r0 = blockIdx.x * TILE_R;

  // ---- Stage price tile [64 x 128] through LDS (coalesced b128, NT) -------
  #pragma unroll
  for (int i = 0; i < (JJ * TILE_R) / (NTHREADS * 4); ++i) {
    const int idx = i * NTHREADS + t;     // float4 index within tile
    const int j   = idx >> 5;             // 32 float4 per j-row
    const int c   = (idx & 31) << 2;      // column (float units)
    const v4f v = __builtin_nontemporal_load(
        (const v4f*)(price + (size_t)j * R + r0 + c));
    float* dst = &lds_price[j * LDS_STRIDE + c];
    dst[0] = v.x; dst[1] = v.y; dst[2] = v.z; dst[3] = v.w;
  }
  __syncthreads();

  const int wave = t >> 5;
  const int lane = t & 31;

  const float b1 = bases[1],  b2 = bases[2];
  const float s1 = scales[1], s2 = scales[2];

  float my_gdp = 0.0f;                  // lane `it` latches row it's outputs
  float my_aec = 0.0f;

  for (int it = 0; it < ROWS_PER_WAVE; ++it) {
    const int rr = wave * ROWS_PER_WAVE + it;
    const int r  = r0 + rr;

    const size_t baseJ = (size_t)r * JJ + 2 * lane;
    const size_t baseM = (size_t)r * MM + lane;

    if (it + 1 < ROWS_PER_WAVE) {       // next-row stream prefetch
      __builtin_prefetch(inventory  + baseJ + JJ, 0, 0);
      __builtin_prefetch(innovation + baseJ + JJ, 0, 0);
      __builtin_prefetch(endowment  + baseJ + JJ, 0, 0);
    }

    // ---- row scalars (wave-uniform addresses) ----
    const float need = need_prev[r];
    float atp = pool_atp[r];
    float adp = pool_adp[r];
    const float amp = pool_amp[r];
    const float gr  = greed[r];
    const float exg = exergy_price[r];

    const float rech = fminf(adp, need * DT_F);
    atp += rech;
    adp -= rech;
    const float aec = (atp + 0.5f * adp) * fast_rcp(atp + adp + amp + EPS_F);

    const float f1 = clampf(b1 + s1 * gr, 0.0f, 0.9f);
    const float f2 = clampf(b2 + s2 * gr, 0.0f, 0.9f);
    const float wealth       = atp;
    const float innov_budget = wealth * f1 * (1.0f / (float)JJ);
    float atp_book           = wealth - wealth * (f1 + f2);

    // ---- per-lane streams (single-use -> non-temporal) ----
    v2f inv  = __builtin_nontemporal_load((const v2f*)(inventory  + baseJ));
    v2f endw = __builtin_nontemporal_load((const v2f*)(endowment  + baseJ));
    v2f innv = __builtin_nontemporal_load((const v2f*)(innovation + baseJ));
    const float qe = __builtin_nontemporal_load(extraction_eff + baseM);
    const float qd = __builtin_nontemporal_load(deposits       + baseM);

    inv.x = fmaxf(inv.x + DT_F * endw.x, 0.0f);
    inv.y = fmaxf(inv.y + DT_F * endw.y, 0.0f);
    const float ib = DT_F * 0.05f * innov_budget;
    innv.x = innv.x + ib * fast_rcp(1.0f + innv.x);
    innv.y = innv.y + ib * fast_rcp(1.0f + innv.y);

    // ---- extraction ----
    const float q    = qe * qd * DT_F;
    const float qsum = wave_sum32(q);
    const float cost = qsum * 0.5f;
    const float scl  = clampf(atp_book * fast_rcp(cost + EPS_F), 0.0f, 1.0f);
    atp_book -= scl * cost;
    const float q_scaled_sum = qsum * scl;

    // ---- production ----
    const float rx = 0.1f * DT_F * innv.x * inv.x * aec;
    const float ry = 0.1f * DT_F * innv.y * inv.y * aec;
    const float rsum   = wave_sum32(rx + ry);
    const float pcost  = rsum * 0.3f;
    const float pscale = clampf(atp_book * fast_rcp(pcost + EPS_F), 0.0f, 1.0f);

    // ---- gdp = pscale * (rate . price_t) + q.sum * exergy_price_new ----
    const float px = lds_price[(2 * lane)     * LDS_STRIDE + rr];
    const float py = lds_price[(2 * lane + 1) * LDS_STRIDE + rr];
    const float dot = wave_sum32(rx * px + ry * py);

    const float exergy = clampf(exg * (1.0f + 0.02f * (0.5f - aec)),
                                1e-4f, 1e4f);
    const float gdp = pscale * dot + q_scaled_sum * exergy;

    // latch this row's outputs into lane `it` (results valid in all lanes)
    const bool mine = (lane == it);
    my_gdp = mine ? gdp : my_gdp;
    my_aec = mine ? aec : my_aec;
  }

  // ---- two coalesced stores per wave (lanes 0..15 own the wave's 16 rows)
  if (lane < ROWS_PER_WAVE) {
    const int r = r0 + wave * ROWS_PER_WAVE + lane;
    out_gdp[r] = my_gdp;
    out_aec[r] = my_aec;
  }
}

extern "C" void kernel_launch(void* const* d_in, const int* in_sizes, int n_in,
                              void* d_out, int out_size, void* d_ws, size_t ws_size,
                              hipStream_t stream) {
  (void)n_in; (void)d_ws; (void)ws_size; (void)out_size;

  const float* need_prev      = (const float*)d_in[0];
  const float* inventory      = (const float*)d_in[1];
  const float* endowment      = (const float*)d_in[2];
  const float* price          = (const float*)d_in[3];
  const float* pool_atp       = (const float*)d_in[4];
  const float* pool_adp       = (const float*)d_in[5];
  const float* pool_amp       = (const float*)d_in[6];
  // d_in[7] population: dead w.r.t. outputs
  const float* greed          = (const float*)d_in[8];
  const float* innovation     = (const float*)d_in[9];
  const float* extraction_eff = (const float*)d_in[10];
  const float* deposits       = (const float*)d_in[11];
  // d_in[12..14] emit_sink, sink_state, sink_price: dead
  const float* exergy_price   = (const float*)d_in[15];
  // d_in[16] storage_cap: dead
  const float* bases          = (const float*)d_in[17];
  const float* scales         = (const float*)d_in[18];

  const int R = in_sizes[0];            // 131072
  float* out_gdp = (float*)d_out;       // [R]
  float* out_aec = (float*)d_out + R;   // [R]

  const int nblocks = R / TILE_R;       // 1024
  econ_gdp_aec_kernel<<<nblocks, NTHREADS, 0, stream>>>(
      need_prev, inventory, endowment, price,
      pool_atp, pool_adp, pool_amp, greed,
      innovation, extraction_eff, deposits, exergy_price,
      bases, scales, out_gdp, out_aec, R);
}